// GraphCNNLayer2_24919400252234
// MI455X (gfx1250) — compile-verified
//
#include <hip/hip_runtime.h>

typedef __attribute__((ext_vector_type(2))) float v2f;
typedef __attribute__((ext_vector_type(8))) float v8f;

#define B_    4
#define N_    1024
#define L_    8
#define FV_   64
#define OUT_  8
#define WROW_ (L_ + 2 * FV_)   // 136

// ---------------------------------------------------------------------------
// Pre-pass: tv1[b][o][j] = sum_f V[b,j,f] * W[o, 8+f]
//           tv2b[b][i][o] = sum_f V[b,i,f] * W[o, 72+f] + bias[o]
// 65536 threads, each a 64-long dot product. V (1 MB) is L2-resident.
// ---------------------------------------------------------------------------
__global__ void __launch_bounds__(256)
precompute_tv(const float* __restrict__ V, const float* __restrict__ W,
              const float* __restrict__ bias,
              float* __restrict__ tv1, float* __restrict__ tv2b) {
    int idx  = blockIdx.x * blockDim.x + threadIdx.x;
    int half = idx >> 15;          // 0: tv1, 1: tv2b
    int r    = idx & 32767;
    int b    = r >> 13;
    if (half == 0) {
        int o = (r >> 10) & 7;
        int j = r & 1023;
        const float* v = V + (b * N_ + j) * FV_;
        const float* w = W + o * WROW_ + L_;
        float s = 0.f;
#pragma unroll
        for (int f = 0; f < FV_; ++f) s += v[f] * w[f];
        tv1[(b * OUT_ + o) * N_ + j] = s;
    } else {
        int i = (r >> 3) & 1023;
        int o = r & 7;
        const float* v = V + (b * N_ + i) * FV_;
        const float* w = W + o * WROW_ + L_ + FV_;
        float s = bias[o];
#pragma unroll
        for (int f = 0; f < FV_; ++f) s += v[f] * w[f];
        tv2b[(b * N_ + i) * OUT_ + o] = s;
    }
}

// ---------------------------------------------------------------------------
// Main pass: out[b,i,o,j] = relu( sum_l Wa[o,l]*A[b,i,l,j] + tv1[b,o,j]
//                                 + tv2b[b,i,o] )
// One wave handles (b, i0=2p, i1=2p+1) and 256 columns (16 tiles of 16).
// Per tile: 4 chained v_wmma_f32_16x16x4_f32 with block-structured A operands
// so D rows 0-7 = i0 results, rows 8-15 = i1 results (all lanes useful).
// ---------------------------------------------------------------------------
__global__ void __launch_bounds__(256)
graphcnn_main(const float* __restrict__ A, const float* __restrict__ W,
              const float* __restrict__ tv1, const float* __restrict__ tv2b,
              float* __restrict__ out) {
    const int lane   = threadIdx.x & 31;
    const int wave   = threadIdx.x >> 5;
    const int g      = blockIdx.x * 8 + wave;   // 0..8191
    const int jchunk = g & 3;                   // 4 chunks of 256 columns
    const int ipair  = (g >> 2) & 511;          // 512 i-pairs per batch
    const int b      = g >> 11;                 // 4 batches
    const int i0 = ipair * 2, i1 = i0 + 1;
    const int m  = lane & 15;                   // row-in-half / column-in-tile
    const int hi = lane >> 4;                   // 0: lanes 0-15, 1: lanes 16-31

    // ---- A-operand fragments (Wa in rows 0-7 or 8-15; other half zero) ----
    // 16x4 f32 A layout: .x -> K=2*hi, .y -> K=2*hi+1, row M = m.
    const int k0 = 2 * hi, k1 = 2 * hi + 1;
    v2f at1, at2, ab1, ab2;
    at1.x = (m < 8) ? W[m * WROW_ + k0]            : 0.f;   // rows 0-7: Wa[:,0:4]
    at1.y = (m < 8) ? W[m * WROW_ + k1]            : 0.f;
    at2.x = (m < 8) ? W[m * WROW_ + 4 + k0]        : 0.f;   // rows 0-7: Wa[:,4:8]
    at2.y = (m < 8) ? W[m * WROW_ + 4 + k1]        : 0.f;
    ab1.x = (m >= 8) ? W[(m - 8) * WROW_ + k0]     : 0.f;   // rows 8-15: Wa[:,0:4]
    ab1.y = (m >= 8) ? W[(m - 8) * WROW_ + k1]     : 0.f;
    ab2.x = (m >= 8) ? W[(m - 8) * WROW_ + 4 + k0] : 0.f;   // rows 8-15: Wa[:,4:8]
    ab2.y = (m >= 8) ? W[(m - 8) * WROW_ + 4 + k1] : 0.f;

    // ---- per-(i,o) correction, selected per half-wave, hoisted ----
    const int isel = hi ? i1 : i0;
    float tsel[OUT_];
#pragma unroll
    for (int r = 0; r < OUT_; ++r)
        tsel[r] = tv2b[(b * N_ + isel) * OUT_ + r];

    const float* A0 = A + ((b * N_ + i0) * L_) * N_;   // 8x1024 slab for i0
    const float* A1 = A + ((b * N_ + i1) * L_) * N_;   // 8x1024 slab for i1
    const float* t1 = tv1 + b * OUT_ * N_;
    float* obase = out + ((b * N_ + isel) * OUT_) * N_;

    const int jbase = jchunk * 256;
#pragma unroll 1
    for (int t = 0; t < 16; ++t) {
        const int c = jbase + t * 16 + m;   // this lane's column

        // B fragments (4x16 f32): .x -> K=2*hi, .y -> K=2*hi+1, col N = m
        v2f b01, b02, b11, b12;
        b01.x = A0[k0 * N_ + c];        b01.y = A0[k1 * N_ + c];        // i0, l 0-3
        b02.x = A0[(4 + k0) * N_ + c];  b02.y = A0[(4 + k1) * N_ + c];  // i0, l 4-7
        b11.x = A1[k0 * N_ + c];        b11.y = A1[k1 * N_ + c];        // i1, l 0-3
        b12.x = A1[(4 + k0) * N_ + c];  b12.y = A1[(4 + k1) * N_ + c];  // i1, l 4-7

        // C init: row r carries tv1[b,r,j] + tv2b[b,i(half),r]
        v8f cc;
#pragma unroll
        for (int r = 0; r < OUT_; ++r)
            cc[r] = t1[r * N_ + c] + tsel[r];

        // D rows 0-7 accumulate i0, rows 8-15 accumulate i1
        cc = __builtin_amdgcn_wmma_f32_16x16x4_f32(false, at1, false, b01,
                                                   (short)0, cc, false, false);
        cc = __builtin_amdgcn_wmma_f32_16x16x4_f32(false, at2, false, b02,
                                                   (short)0, cc, false, false);
        cc = __builtin_amdgcn_wmma_f32_16x16x4_f32(false, ab1, false, b11,
                                                   (short)0, cc, false, false);
        cc = __builtin_amdgcn_wmma_f32_16x16x4_f32(false, ab2, false, b12,
                                                   (short)0, cc, false, false);

        // ReLU + store: VGPR r = (o=r, i0) on lanes 0-15, (o=r, i1) on 16-31
#pragma unroll
        for (int r = 0; r < OUT_; ++r) {
            float v = cc[r];
            v = v > 0.f ? v : 0.f;
            obase[r * N_ + c] = v;
        }
    }
}

extern "C" void kernel_launch(void* const* d_in, const int* in_sizes, int n_in,
                              void* d_out, int out_size, void* d_ws, size_t ws_size,
                              hipStream_t stream) {
    const float* A    = (const float*)d_in[0];
    const float* V    = (const float*)d_in[1];
    const float* W    = (const float*)d_in[2];
    const float* bias = (const float*)d_in[3];
    float* out  = (float*)d_out;
    float* tv1  = (float*)d_ws;                 // 4*8*1024 floats
    float* tv2b = tv1 + B_ * OUT_ * N_;         // 4*1024*8 floats (256 KB total)

    precompute_tv<<<(2 * 32768) / 256, 256, 0, stream>>>(V, W, bias, tv1, tv2b);
    graphcnn_main<<<1024, 256, 0, stream>>>(A, W, tv1, tv2b, out);
}